// VectorizedConstantVelocityModel_69887707841088
// MI455X (gfx1250) — compile-verified
//
#include <hip/hip_runtime.h>
#include <math.h>

// CDNA5 (gfx1250) wave32. One wave per 16x16 tile of the NxN pair matrix.
// V_WMMA_F32_16X16X4_F32 computes the K=4 Gram cross-terms (z.z, v.v, z.v+v.z);
// the transcendental survival integral (exp + 2x erf) runs on the VALU.

typedef __attribute__((ext_vector_type(2))) float v2f;
typedef __attribute__((ext_vector_type(8))) float v8f;

#define BLOCK 256

__device__ __forceinline__ float blockReduceSum(float v, float* sdata) {
    const int t = threadIdx.x;
    sdata[t] = v;
    __syncthreads();
    for (int s = BLOCK >> 1; s > 0; s >>= 1) {
        if (t < s) sdata[t] += sdata[t + s];
        __syncthreads();
    }
    return sdata[0];
}

// ---------------------------------------------------------------------------
// Event intensity: sum_m ( beta - || (z_i + v_i t) - (z_j + v_j t) ||^2 )
// ---------------------------------------------------------------------------
__global__ void event_kernel(const float* __restrict__ data,
                             const float* __restrict__ z0,
                             const float* __restrict__ v0,
                             const float* __restrict__ beta,
                             float* __restrict__ partials, int M) {
    __shared__ float sdata[BLOCK];
    const float b = beta[0];
    float sum = 0.0f;
    for (int m = blockIdx.x * BLOCK + threadIdx.x; m < M; m += gridDim.x * BLOCK) {
        const int   i = (int)floorf(data[3 * m + 0]);
        const int   j = (int)floorf(data[3 * m + 1]);
        const float t = data[3 * m + 2];
        const float zix = fmaf(v0[2 * i + 0], t, z0[2 * i + 0]);
        const float ziy = fmaf(v0[2 * i + 1], t, z0[2 * i + 1]);
        const float zjx = fmaf(v0[2 * j + 0], t, z0[2 * j + 0]);
        const float zjy = fmaf(v0[2 * j + 1], t, z0[2 * j + 1]);
        const float dx = zix - zjx, dy = ziy - zjy;
        sum += b - (dx * dx + dy * dy);
    }
    const float bs = blockReduceSum(sum, sdata);
    if (threadIdx.x == 0) partials[blockIdx.x] = bs;
}

// ---------------------------------------------------------------------------
// Non-event intensity: one wave32 per 16x16 tile; upper-triangular tiles only.
// WMMA 16x16x4 f32 produces zz = z_i.z_j, vv = v_i.v_j, zv = z_i.v_j + v_i.z_j.
// ---------------------------------------------------------------------------
__global__ void pair_kernel(const float* __restrict__ z0,
                            const float* __restrict__ v0,
                            const float* __restrict__ beta,
                            const float* __restrict__ t0p,
                            const float* __restrict__ tnp,
                            float* __restrict__ partials, int N, int nt) {
    __shared__ float sdata[BLOCK];
    const int wave = (blockIdx.x * BLOCK + threadIdx.x) >> 5;  // wave-uniform
    const int lane = threadIdx.x & 31;
    float sum = 0.0f;

    if (wave < nt * nt) {
        const int bi = wave / nt;
        const int bj = wave - bi * nt;
        if (bj >= bi) {  // wave-uniform branch: EXEC stays all-ones for WMMA
            const float b  = beta[0];
            const float t0 = t0p[0];
            const float tn = tnp[0];
            const int i0 = bi * 16;
            const int j0 = bj * 16;
            const int half = lane >> 4;  // 0: lanes 0-15, 1: lanes 16-31
            const int l16  = lane & 15;

            // --- A fragment: 16x4 f32. VGPR0 = K(0|2), VGPR1 = K(1|3) per half.
            int ra = i0 + l16; if (ra > N - 1) ra = N - 1;
            v2f A;
            if (half == 0) { A.x = z0[2 * ra + 0]; A.y = z0[2 * ra + 1]; }
            else           { A.x = v0[2 * ra + 0]; A.y = v0[2 * ra + 1]; }

            // --- B fragments: 4x16 f32, column j0 + l16; half selects K rows.
            int cb = j0 + l16; if (cb > N - 1) cb = N - 1;
            const float zjx = z0[2 * cb + 0], zjy = z0[2 * cb + 1];
            const float vjx = v0[2 * cb + 0], vjy = v0[2 * cb + 1];
            v2f B1, B2, B3;
            if (half == 0) { B1.x = zjx; B1.y = zjy;  B2.x = 0.f; B2.y = 0.f;  B3.x = vjx; B3.y = vjy; }
            else           { B1.x = 0.f; B1.y = 0.f;  B2.x = vjx; B2.y = vjy;  B3.x = zjx; B3.y = zjy; }

            v8f zz = {}; v8f vv = {}; v8f zv = {};
            // (neg_a, A, neg_b, B, c_mod, C, reuse_a, reuse_b)
            zz = __builtin_amdgcn_wmma_f32_16x16x4_f32(false, A, false, B1, (short)0, zz, false, false);
            vv = __builtin_amdgcn_wmma_f32_16x16x4_f32(false, A, false, B2, (short)0, vv, false, false);
            zv = __builtin_amdgcn_wmma_f32_16x16x4_f32(false, A, false, B3, (short)0, zv, false, false);

            // j-side per-node scalars (column = l16 for both halves)
            const float nzj = zjx * zjx + zjy * zjy;
            const float nvj = vjx * vjx + vjy * vjy;
            const float zvj = zjx * vjx + zjy * vjy;
            const int jn = j0 + l16;

            const float SQRT_PI_OVER_2 = 0.8862269254527580f;

            // C/D layout: VGPR r, lanes 0-15 -> M=r; lanes 16-31 -> M=r+8.
            #pragma unroll
            for (int r = 0; r < 8; ++r) {
                const int ii = i0 + r + 8 * half;
                if (ii < N && jn < N && jn > ii) {  // strict upper triangle
                    const float zix = z0[2 * ii + 0], ziy = z0[2 * ii + 1];
                    const float vix = v0[2 * ii + 0], viy = v0[2 * ii + 1];
                    const float nzi = zix * zix + ziy * ziy;
                    const float nvi = vix * vix + viy * viy;
                    const float zvi = zix * vix + ziy * viy;

                    const float aa  = nvi + nvj - 2.0f * vv[r];           // |dv|^2
                    const float cc  = nzi + nzj - 2.0f * zz[r];           // |dz|^2
                    const float bbv = 2.0f * (zvi + zvj - zv[r]);         // 2 dz.dv

                    const float a_safe = fmaxf(aa, 1e-10f);
                    const float sqa    = sqrtf(a_safe);
                    const float pref   = SQRT_PI_OVER_2 / sqa *
                                         expf(b + bbv * bbv / (4.0f * a_safe) - cc);
                    const float off    = bbv / (2.0f * sqa);
                    const float upper  = erff(fmaf(sqa, tn, off));
                    const float lower  = erff(fmaf(sqa, t0, off));
                    sum += pref * (upper - lower);
                }
            }
        }
    }
    const float bs = blockReduceSum(sum, sdata);
    if (threadIdx.x == 0) partials[blockIdx.x] = bs;
}

// ---------------------------------------------------------------------------
// Final: out = event_sum - nonevent_sum
// ---------------------------------------------------------------------------
__global__ void finalize_kernel(const float* __restrict__ pairPartials, int nPair,
                                const float* __restrict__ evtPartials, int nEvt,
                                float* __restrict__ out) {
    __shared__ float sdata[BLOCK];
    float s = 0.0f;
    for (int k = threadIdx.x; k < nEvt; k += BLOCK) s += evtPartials[k];
    for (int k = threadIdx.x; k < nPair; k += BLOCK) s -= pairPartials[k];
    const float tot = blockReduceSum(s, sdata);
    if (threadIdx.x == 0) out[0] = tot;
}

extern "C" void kernel_launch(void* const* d_in, const int* in_sizes, int n_in,
                              void* d_out, int out_size, void* d_ws, size_t ws_size,
                              hipStream_t stream) {
    // setup_inputs order: data, t0, tn, beta, z0, v0
    const float* data = (const float*)d_in[0];
    const float* t0p  = (const float*)d_in[1];
    const float* tnp  = (const float*)d_in[2];
    const float* beta = (const float*)d_in[3];
    const float* z0   = (const float*)d_in[4];
    const float* v0   = (const float*)d_in[5];

    const int M  = in_sizes[0] / 3;
    const int N  = in_sizes[4] / 2;
    const int nt = (N + 15) / 16;                     // 16x16 tiles per side
    const int pairWaves  = nt * nt;                   // one wave32 per tile
    const int pairBlocks = (pairWaves * 32 + BLOCK - 1) / BLOCK;
    const int evtBlocks  = 256;

    float* ws           = (float*)d_ws;
    float* pairPartials = ws;
    float* evtPartials  = ws + pairBlocks;

    event_kernel<<<evtBlocks, BLOCK, 0, stream>>>(data, z0, v0, beta, evtPartials, M);
    pair_kernel<<<pairBlocks, BLOCK, 0, stream>>>(z0, v0, beta, t0p, tnp,
                                                  pairPartials, N, nt);
    finalize_kernel<<<1, BLOCK, 0, stream>>>(pairPartials, pairBlocks,
                                             evtPartials, evtBlocks, (float*)d_out);
}